// PGA_42923903156739
// MI455X (gfx1250) — compile-verified
//
#include <hip/hip_runtime.h>

typedef __attribute__((ext_vector_type(2))) float v2f;
typedef __attribute__((ext_vector_type(8))) float v8f;

#define M_DIM      64
#define N_DIM      2048
#define K_DIM      512
#define MAX_ITERS  300
#define TOL2       1e-8f      // (1e-4)^2, compare squared norms
#define CLAMP_MIN  1e-8f
#define NCOL       32         // columns of H owned by one workgroup
#define HSTRIDE    68         // padded row length of transposed H slice

// ---------------------------------------------------------------------------
// Kernel 1: step = 1.995 / (||W||_F^2 + Lambda); stash step and Lambda in ws.
// ---------------------------------------------------------------------------
__global__ void __launch_bounds__(256)
k_step(const float* __restrict__ W, const float* __restrict__ Lambda,
       float* __restrict__ wsf) {
    __shared__ float red[8];
    float s = 0.f;
    for (int i = threadIdx.x; i < K_DIM * M_DIM; i += 256) {
        float w = W[i];
        s += w * w;
    }
    for (int off = 16; off > 0; off >>= 1) s += __shfl_xor(s, off, 32);
    int wave = threadIdx.x >> 5, lane = threadIdx.x & 31;
    if (lane == 0) red[wave] = s;
    __syncthreads();
    if (threadIdx.x == 0) {
        float t = 0.f;
        for (int i = 0; i < 8; ++i) t += red[i];
        wsf[0] = 1.995f / (t + Lambda[0]);
        wsf[1] = Lambda[0];
    }
}

// ---------------------------------------------------------------------------
// Kernel 2: M = I - step*(W^T W + lambda I)   (64x64, K=512), via f32 WMMA.
// One workgroup, 8 waves, 2 tiles (16x16) per wave.
// ---------------------------------------------------------------------------
__global__ void __launch_bounds__(256)
k_buildM(const float* __restrict__ W, float* __restrict__ wsf) {
    const float step = wsf[0];
    const float lam  = wsf[1];
    float* Mout = wsf + 64;
    const int tid  = threadIdx.x;
    const int wave = tid >> 5;
    const int lane = tid & 31;
    const int l  = lane & 15;
    const int hi = lane >> 4;

    for (int t = wave * 2; t < wave * 2 + 2; ++t) {
        const int mt = t >> 2;        // row tile of output
        const int nt = t & 3;         // col tile of output
        v8f acc = {};
        for (int k = 0; k < K_DIM; k += 4) {
            const int kk = k + 2 * hi;
            v2f a, b;
            // A = W^T tile: A[m][k] = W[k][m]
            a[0] = W[(kk + 0) * M_DIM + mt * 16 + l];
            a[1] = W[(kk + 1) * M_DIM + mt * 16 + l];
            // B = W tile: B[k][n] = W[k][n]
            b[0] = W[(kk + 0) * M_DIM + nt * 16 + l];
            b[1] = W[(kk + 1) * M_DIM + nt * 16 + l];
            acc = __builtin_amdgcn_wmma_f32_16x16x4_f32(
                false, a, false, b, (short)0, acc, false, false);
        }
        for (int v = 0; v < 8; ++v) {
            const int row = mt * 16 + v + 8 * hi;
            const int col = nt * 16 + l;
            const float diag = (row == col) ? (1.0f - step * lam) : 0.0f;
            Mout[row * M_DIM + col] = diag - step * acc[v];
        }
    }
}

// ---------------------------------------------------------------------------
// Kernel 3: Bs = step * (W^T V)  (64x2048, K=512), staged into d_out.
// 64 blocks x 8 waves = 512 waves; one 16x16 output tile per wave.
// ---------------------------------------------------------------------------
__global__ void __launch_bounds__(256)
k_buildB(const float* __restrict__ W, const float* __restrict__ V,
         const float* __restrict__ wsf, float* __restrict__ Bs) {
    const float step = wsf[0];
    const int tid  = threadIdx.x;
    const int wave = tid >> 5;
    const int lane = tid & 31;
    const int l  = lane & 15;
    const int hi = lane >> 4;
    const int w  = blockIdx.x * 8 + wave;   // 0..511
    const int mt = w & 3;                   // row tile 0..3
    const int ct = w >> 2;                  // col tile 0..127

    v8f acc = {};
    for (int k = 0; k < K_DIM; k += 4) {
        const int kk = k + 2 * hi;
        v2f a, b;
        a[0] = W[(kk + 0) * M_DIM + mt * 16 + l];
        a[1] = W[(kk + 1) * M_DIM + mt * 16 + l];
        b[0] = V[(size_t)(kk + 0) * N_DIM + ct * 16 + l];
        b[1] = V[(size_t)(kk + 1) * N_DIM + ct * 16 + l];
        acc = __builtin_amdgcn_wmma_f32_16x16x4_f32(
            false, a, false, b, (short)0, acc, false, false);
    }
    for (int v = 0; v < 8; ++v) {
        const int row = mt * 16 + v + 8 * hi;
        const int col = ct * 16 + l;
        Bs[row * N_DIM + col] = step * acc[v];
    }
}

// ---------------------------------------------------------------------------
// Kernel 4: persistent per-slice iteration.
//   H_slice <- max(M * H_slice + Bs_slice, 1e-8)
// All loop-invariants (M operands, Bs tile, H_old tile) live in registers;
// LDS holds only the transposed, double-buffered H slice exchanged between
// waves. One barrier per iteration (red[] double-buffered).
// 64 blocks (32 columns each) x 256 threads (8 waves, one 16x16 tile each).
// ---------------------------------------------------------------------------
__global__ void __launch_bounds__(256)
k_iterate(const float* __restrict__ H0, const float* __restrict__ wsf,
          float* __restrict__ out /* holds Bs on entry, final H on exit */) {
    __shared__ float sHT[2][NCOL][HSTRIDE];   // transposed: [buf][col][row(=K)]
    __shared__ float red[2][16];

    const float* Mg = wsf + 64;
    const int tid  = threadIdx.x;
    const int col0 = blockIdx.x * NCOL;
    const int wave = tid >> 5;
    const int lane = tid & 31;
    const int l  = lane & 15;
    const int hi = lane >> 4;
    const int mt = wave >> 1;            // row tile 0..3
    const int nt = wave & 1;             // col tile 0..1
    const int r0 = mt * 16 + 8 * hi;     // first of this lane's 8 rows (mult of 8)
    const int c  = nt * 16 + l;          // this lane's column within the slice

    // A operands (rows of M): iteration-invariant, keep all 16 K-steps in regs.
    v2f areg[16];
#pragma unroll
    for (int ks = 0; ks < 16; ++ks) {
        const int kk = 4 * ks + 2 * hi;
        areg[ks][0] = Mg[(mt * 16 + l) * M_DIM + kk + 0];
        areg[ks][1] = Mg[(mt * 16 + l) * M_DIM + kk + 1];
    }

    // Bs tile (accumulator seed) and current H tile: registers, C/D layout.
    v8f bs, hcur;
#pragma unroll
    for (int v = 0; v < 8; ++v) {
        bs[v]   = out[(size_t)(r0 + v) * N_DIM + col0 + c];
        hcur[v] = H0 [(size_t)(r0 + v) * N_DIM + col0 + c];
        sHT[0][c][r0 + v] = hcur[v];
    }
    __syncthreads();

    int cur = 0;
    for (int it = 0; it < MAX_ITERS; ++it) {
        const int nxt = cur ^ 1;
        const int p   = it & 1;

        v8f acc = bs;
#pragma unroll
        for (int ks = 0; ks < 16; ++ks) {
            const int kk = 4 * ks + 2 * hi;
            const v2f b = *(const v2f*)&sHT[cur][c][kk];   // b64, conflict-free
            acc = __builtin_amdgcn_wmma_f32_16x16x4_f32(
                false, areg[ks], false, b, (short)0, acc, false, false);
        }

        float d2 = 0.f, o2 = 0.f;
        v8f hn;
#pragma unroll
        for (int v = 0; v < 8; ++v) {
            const float x  = acc[v] < CLAMP_MIN ? CLAMP_MIN : acc[v];
            const float df = x - hcur[v];
            d2 += df * df;
            o2 += hcur[v] * hcur[v];
            hn[v] = x;
        }
        *(v8f*)&sHT[nxt][c][r0] = hn;   // contiguous 8 floats, wide ds stores
        hcur = hn;

        for (int off = 16; off > 0; off >>= 1) {
            d2 += __shfl_xor(d2, off, 32);
            o2 += __shfl_xor(o2, off, 32);
        }
        if (lane == 0) { red[p][wave] = d2; red[p][8 + wave] = o2; }
        __syncthreads();

        float D2 = 0.f, O2 = 0.f;
#pragma unroll
        for (int i = 0; i < 8; ++i) { D2 += red[p][i]; O2 += red[p][8 + i]; }

        cur = nxt;
        const int nit = it + 1;
        // reference: continue while (nit < 300) && (nit <= 1 || ||dH|| >= ||H||*TOL)
        if (!((nit < MAX_ITERS) && ((nit <= 1) || (D2 >= O2 * TOL2))))
            break;
    }

    // Final H tile lives in registers; write straight to global.
#pragma unroll
    for (int v = 0; v < 8; ++v)
        out[(size_t)(r0 + v) * N_DIM + col0 + c] = hcur[v];
}

// ---------------------------------------------------------------------------
extern "C" void kernel_launch(void* const* d_in, const int* in_sizes, int n_in,
                              void* d_out, int out_size, void* d_ws, size_t ws_size,
                              hipStream_t stream) {
    const float* V   = (const float*)d_in[0];   // 512 x 2048
    const float* H   = (const float*)d_in[1];   // 64 x 2048
    const float* W   = (const float*)d_in[2];   // 512 x 64
    const float* Lam = (const float*)d_in[3];   // scalar
    float* out = (float*)d_out;                 // 64 x 2048
    float* wsf = (float*)d_ws;                  // [0]=step, [1]=lambda, +64: M(64x64)

    k_step   <<<1,  256, 0, stream>>>(W, Lam, wsf);
    k_buildM <<<1,  256, 0, stream>>>(W, wsf);
    k_buildB <<<64, 256, 0, stream>>>(W, V, wsf, out);
    k_iterate<<<64, 256, 0, stream>>>(H, wsf, out);
}